// MemoryBlock_69552700391763
// MI455X (gfx1250) — compile-verified
//
#include <hip/hip_runtime.h>

typedef __attribute__((ext_vector_type(2))) float v2f;
typedef __attribute__((ext_vector_type(8))) float v8f;

#define KB_PER_BLOCK 2048   // K elements handled per block (grid covers all K)
#define KC 64               // LDS-staged K chunk
#define LSTR 68             // LDS row stride (floats): keeps float4 alignment, spreads banks

// ---------------------------------------------------------------------------
// Kernel 0: zero the reduction workspace
// ---------------------------------------------------------------------------
__global__ void zero_ws_kernel(float* ws, int n) {
    int i = blockIdx.x * blockDim.x + threadIdx.x;
    if (i < n) ws[i] = 0.0f;
}

// ---------------------------------------------------------------------------
// Kernel 1: per-level streaming pass.
//   wsFsq[b]  += inv_n * sum_k f[b,k]^2
//   wsMsq[m]  += inv_n * sum_k m[m,k]^2
//   wsCross[b*30+m] += -2*inv_n * sum_k f[b,k]*m[m,k]   (via V_WMMA_F32_16X16X4_F32)
// Block = 128 threads = 4 waves; wave w owns output tile (bt = w&1, mt = w>>1).
// ---------------------------------------------------------------------------
__global__ __launch_bounds__(128) void cross_norms_kernel(
    const float* __restrict__ f, const float* __restrict__ m, int N, float inv_n,
    float* __restrict__ wsCross, float* __restrict__ wsFsq, float* __restrict__ wsMsq)
{
    __shared__ float fT[32 * LSTR];
    __shared__ float mT[32 * LSTR];
    __shared__ float red[128];

    const int tid   = threadIdx.x;
    const int row   = tid >> 2;      // 0..31 : row this thread stages
    const int seg   = tid & 3;       // 0..3  : 16-float segment within row
    const int lane  = tid & 31;
    const int wave  = tid >> 5;      // 0..3
    const int bt    = wave & 1;      // batch-row tile (0: rows 0-15, 1: rows 16-31)
    const int mt    = wave >> 1;     // memory-col tile
    const int r16   = lane & 15;
    const int khalf = lane >> 4;     // 0/1 : which K-pair half (f32 16x4 fragment layout)

    const int kbase = blockIdx.x * KB_PER_BLOCK;

    v8f  acc = {0.f, 0.f, 0.f, 0.f, 0.f, 0.f, 0.f, 0.f};
    float fsq = 0.f, msq = 0.f;

    for (int kc = kbase; kc < kbase + KB_PER_BLOCK; kc += KC) {
        // ---- stage f tile [32 x 64] (coalesced b128 loads) ----
        {
            const float4* src = (const float4*)(f + (size_t)row * N + kc + seg * 16);
            float4 a = src[0], b = src[1], c = src[2], d = src[3];
            float4* dst = (float4*)&fT[row * LSTR + seg * 16];
            dst[0] = a; dst[1] = b; dst[2] = c; dst[3] = d;
            fsq += a.x*a.x + a.y*a.y + a.z*a.z + a.w*a.w;
            fsq += b.x*b.x + b.y*b.y + b.z*b.z + b.w*b.w;
            fsq += c.x*c.x + c.y*c.y + c.z*c.z + c.w*c.w;
            fsq += d.x*d.x + d.y*d.y + d.z*d.z + d.w*d.w;
        }
        // ---- stage m tile [30 x 64], rows 30/31 zero-padded ----
        {
            float4* dst = (float4*)&mT[row * LSTR + seg * 16];
            if (row < 30) {
                const float4* src = (const float4*)(m + (size_t)row * N + kc + seg * 16);
                float4 a = src[0], b = src[1], c = src[2], d = src[3];
                dst[0] = a; dst[1] = b; dst[2] = c; dst[3] = d;
                msq += a.x*a.x + a.y*a.y + a.z*a.z + a.w*a.w;
                msq += b.x*b.x + b.y*b.y + b.z*b.z + b.w*b.w;
                msq += c.x*c.x + c.y*c.y + c.z*c.z + c.w*c.w;
                msq += d.x*d.x + d.y*d.y + d.z*d.z + d.w*d.w;
            } else {
                float4 z = make_float4(0.f, 0.f, 0.f, 0.f);
                dst[0] = z; dst[1] = z; dst[2] = z; dst[3] = z;
            }
        }
        __syncthreads();

        // ---- 16 x V_WMMA_F32_16X16X4_F32 over the 64-wide chunk ----
        const float* fbase = &fT[(bt * 16 + r16) * LSTR + 2 * khalf];
        const float* mbase = &mT[(mt * 16 + r16) * LSTR + 2 * khalf];
        #pragma unroll
        for (int k0 = 0; k0 < KC; k0 += 4) {
            v2f av = *(const v2f*)(fbase + k0);   // A: 16x4 fp32 fragment (2 VGPRs)
            v2f bv = *(const v2f*)(mbase + k0);   // B: 4x16 fp32 fragment (2 VGPRs)
            acc = __builtin_amdgcn_wmma_f32_16x16x4_f32(
                false, av, false, bv, (short)0, acc, false, false);
        }
        __syncthreads();
    }

    // ---- row-norm reductions (4 stagers per row) ----
    red[tid] = fsq;
    __syncthreads();
    if (seg == 0) {
        float s = red[tid] + red[tid + 1] + red[tid + 2] + red[tid + 3];
        atomicAdd(&wsFsq[row], s * inv_n);
    }
    __syncthreads();
    red[tid] = msq;
    __syncthreads();
    if (seg == 0 && row < 30) {
        float s = red[tid] + red[tid + 1] + red[tid + 2] + red[tid + 3];
        atomicAdd(&wsMsq[row], s * inv_n);
    }

    // ---- cross writeback: D-tile layout: VGPR j -> M = j + 8*(lane>=16), N = lane&15 ----
    const int gm = mt * 16 + r16;
    if (gm < 30) {
        const float scale = -2.0f * inv_n;
        #pragma unroll
        for (int j = 0; j < 8; ++j) {
            int gb = bt * 16 + j + 8 * khalf;
            atomicAdd(&wsCross[gb * 30 + gm], acc[j] * scale);
        }
    }
}

// ---------------------------------------------------------------------------
// Kernel 2: argmin over 30 memory samples per batch element
// ---------------------------------------------------------------------------
__global__ void argmin_kernel(const float* __restrict__ wsCross,
                              const float* __restrict__ wsFsq,
                              const float* __restrict__ wsMsq,
                              int* __restrict__ sel)
{
    int b = threadIdx.x;
    if (b < 32) {
        float fb = wsFsq[b];
        float best = 3.4e38f;
        int bi = 0;
        for (int mm = 0; mm < 30; ++mm) {
            float d = fb + wsMsq[mm] + wsCross[b * 30 + mm];
            if (d < best) { best = d; bi = mm; }
        }
        sel[b] = bi;
    }
}

// ---------------------------------------------------------------------------
// Kernel 3: out_l = concat([f, (mem[sel[b]] - f)^2], axis=C), all float4.
// All per-level constants are powers of two -> pure shift/mask index decode.
// ---------------------------------------------------------------------------
template <int C, int PLANEQ>
__device__ __forceinline__ void emit_level(const float4* __restrict__ f,
                                           const float4* __restrict__ m,
                                           const int* __restrict__ sel,
                                           float4* __restrict__ out, int local)
{
    const int PER_B = 2 * C * PLANEQ;            // power of two
    int b   = local / PER_B;
    int rem = local - b * PER_B;
    int c   = rem / PLANEQ;
    int p   = rem - c * PLANEQ;
    if (c < C) {
        out[local] = f[(b * C + c) * PLANEQ + p];
    } else {
        int c2 = c - C;
        float4 fv = f[(b * C + c2) * PLANEQ + p];
        float4 mv = m[(sel[b] * C + c2) * PLANEQ + p];
        float4 d;
        d.x = (mv.x - fv.x) * (mv.x - fv.x);
        d.y = (mv.y - fv.y) * (mv.y - fv.y);
        d.z = (mv.z - fv.z) * (mv.z - fv.z);
        d.w = (mv.w - fv.w) * (mv.w - fv.w);
        out[local] = d;
    }
}

__global__ __launch_bounds__(256) void build_out_kernel(
    const float* __restrict__ f0, const float* __restrict__ f1, const float* __restrict__ f2,
    const float* __restrict__ m0, const float* __restrict__ m1, const float* __restrict__ m2,
    const int* __restrict__ sel, float4* __restrict__ out)
{
    const int L0 = 32 * 128 * 1024;  // level-0 out size in float4
    const int L1 = 32 * 256 * 256;   // level-1
    const int L2 = 32 * 512 * 64;    // level-2
    int i = blockIdx.x * 256 + threadIdx.x;
    if (i < L0) {
        emit_level<64, 1024>((const float4*)f0, (const float4*)m0, sel, out, i);
    } else if (i < L0 + L1) {
        emit_level<128, 256>((const float4*)f1, (const float4*)m1, sel, out + L0, i - L0);
    } else if (i < L0 + L1 + L2) {
        emit_level<256, 64>((const float4*)f2, (const float4*)m2, sel, out + L0 + L1,
                            i - L0 - L1);
    }
}

// ---------------------------------------------------------------------------
extern "C" void kernel_launch(void* const* d_in, const int* in_sizes, int n_in,
                              void* d_out, int out_size, void* d_ws, size_t ws_size,
                              hipStream_t stream) {
    const float* f0 = (const float*)d_in[0];
    const float* f1 = (const float*)d_in[1];
    const float* f2 = (const float*)d_in[2];
    const float* m0 = (const float*)d_in[3];
    const float* m1 = (const float*)d_in[4];
    const float* m2 = (const float*)d_in[5];

    float* wsf     = (float*)d_ws;
    float* wsCross = wsf;           // 32*30 = 960 floats
    float* wsFsq   = wsf + 960;     // 32
    float* wsMsq   = wsf + 992;     // 30
    int*   sel     = (int*)(wsf + 1024); // 32 ints

    // zero the accumulators (cross + fsq + msq = 1022 floats)
    zero_ws_kernel<<<1, 1024, 0, stream>>>(wsf, 1022);

    const int N0 = 64 * 64 * 64;    // 262144
    const int N1 = 128 * 32 * 32;   // 131072
    const int N2 = 256 * 16 * 16;   // 65536

    cross_norms_kernel<<<N0 / KB_PER_BLOCK, 128, 0, stream>>>(
        f0, m0, N0, 1.0f / (float)N0, wsCross, wsFsq, wsMsq);
    cross_norms_kernel<<<N1 / KB_PER_BLOCK, 128, 0, stream>>>(
        f1, m1, N1, 1.0f / (float)N1, wsCross, wsFsq, wsMsq);
    cross_norms_kernel<<<N2 / KB_PER_BLOCK, 128, 0, stream>>>(
        f2, m2, N2, 1.0f / (float)N2, wsCross, wsFsq, wsMsq);

    argmin_kernel<<<1, 32, 0, stream>>>(wsCross, wsFsq, wsMsq, sel);

    const int total4 = 32 * 128 * 1024 + 32 * 256 * 256 + 32 * 512 * 64; // 7340032
    build_out_kernel<<<(total4 + 255) / 256, 256, 0, stream>>>(
        f0, f1, f2, m0, m1, m2, sel, (float4*)d_out);
}